// GCN_63161789055109
// MI455X (gfx1250) — compile-verified
//
#include <hip/hip_runtime.h>

#define NN   100000   // nodes
#define NE   3200000  // edges
#define FIN  512
#define FHID 32
#define NCLS 16

typedef __attribute__((ext_vector_type(2))) float v2f;
typedef __attribute__((ext_vector_type(4))) float v4f;
typedef __attribute__((ext_vector_type(8))) float v8f;

// ---------------------------------------------------------------- utilities
__global__ __launch_bounds__(256) void zero_ws(float* __restrict__ p, int n4) {
  int i = blockIdx.x * 256 + threadIdx.x;     // n4 = count of float4s
  if (i < n4) ((v4f*)p)[i] = (v4f){0.f, 0.f, 0.f, 0.f};
}

__global__ __launch_bounds__(256) void degree_kernel(const int* __restrict__ dst,
                                                     float* __restrict__ deg) {
  int e = blockIdx.x * 256 + threadIdx.x;
  if (e < NE) unsafeAtomicAdd(&deg[dst[e]], 1.0f);
}

__global__ __launch_bounds__(256) void dinv_kernel(float* __restrict__ deg_dinv) {
  int i = blockIdx.x * 256 + threadIdx.x;
  if (i < NN) deg_dinv[i] = rsqrtf(deg_dinv[i] + 1.0f);  // self-loop: deg+1
}

// ---------------------------------------------------- GEMM1: H1 = X @ W1 (WMMA f32)
// block = 256 threads = 8 waves; wave w owns M-tile (blockIdx.x*8 + w) of 16 rows.
// Software-pipelined: while WMMAs consume chunk kc from LDS, chunk kc+32 streams
// from global into registers (branchless row-clamped addressing -> batched loads).
__global__ __launch_bounds__(256) void gemm1_wmma(const float* __restrict__ X,
                                                  const float* __restrict__ W1,
                                                  float* __restrict__ H1) {
  __shared__ float xc[128 * 36];   // X chunk [128 rows][32 k], row stride 36 floats
  __shared__ float w1c[32 * 36];   // W1 chunk transposed: w1c[n*36+kk] = W1[kc+kk][n]
  const int t    = threadIdx.x;
  const int wave = t >> 5;
  const int lane = t & 31;
  const int m    = lane & 15;      // row within tile (A) / col within N-tile (B,C)
  const int kh   = lane >> 4;      // K half-select per ISA 16x4 f32 A layout
  const int rowblk = blockIdx.x * 128;
  const int myrow0 = rowblk + wave * 16;
  const bool active = (myrow0 < NN);

  // per-thread staging coordinates (kc-invariant)
  int srow[4], scol[4];
  const float* xbase[4];
#pragma unroll
  for (int i = 0; i < 4; ++i) {
    int li = i * 256 + t;                 // 0..1023
    srow[i] = li >> 3;                    // 0..127
    scol[i] = (li & 7) * 4;               // 0..28
    int grow = rowblk + srow[i];
    grow = (grow < NN) ? grow : (NN - 1); // clamp: branchless, dead tiles unused
    xbase[i] = X + (size_t)grow * FIN + scol[i];
  }
  const int wkk = (t * 4) >> 5;           // 0..31
  const int wn  = (t * 4) & 31;           // 0,4,...,28

  // prologue: fetch chunk 0 into registers
  v4f xr[4], wr;
#pragma unroll
  for (int i = 0; i < 4; ++i) xr[i] = *(const v4f*)(xbase[i]);
  wr = *(const v4f*)(W1 + (size_t)wkk * FHID + wn);

  v8f acc0 = {0.f,0.f,0.f,0.f,0.f,0.f,0.f,0.f};
  v8f acc1 = {0.f,0.f,0.f,0.f,0.f,0.f,0.f,0.f};

  for (int kc = 0; kc < FIN; kc += 32) {
    __syncthreads();                      // prior chunk's LDS reads complete
    // commit current chunk registers -> LDS
#pragma unroll
    for (int i = 0; i < 4; ++i) *(v4f*)(&xc[srow[i] * 36 + scol[i]]) = xr[i];
    w1c[(wn + 0) * 36 + wkk] = wr.x;
    w1c[(wn + 1) * 36 + wkk] = wr.y;
    w1c[(wn + 2) * 36 + wkk] = wr.z;
    w1c[(wn + 3) * 36 + wkk] = wr.w;
    __syncthreads();

    // prefetch next chunk (in flight during WMMA compute below)
    int kn = kc + 32;
    if (kn < FIN) {
#pragma unroll
      for (int i = 0; i < 4; ++i) xr[i] = *(const v4f*)(xbase[i] + kn);
      wr = *(const v4f*)(W1 + (size_t)(kn + wkk) * FHID + wn);
    }

    if (active) {                         // wave-uniform: EXEC all-ones inside
#pragma unroll
      for (int ks = 0; ks < 8; ++ks) {
        int kk0 = ks * 4 + 2 * kh;
        v2f a  = *(const v2f*)(&xc[(wave * 16 + m) * 36 + kk0]);
        v2f b0 = *(const v2f*)(&w1c[(m     ) * 36 + kk0]);
        v2f b1 = *(const v2f*)(&w1c[(m + 16) * 36 + kk0]);
        acc0 = __builtin_amdgcn_wmma_f32_16x16x4_f32(false, a, false, b0,
                                                     (short)0, acc0, false, false);
        acc1 = __builtin_amdgcn_wmma_f32_16x16x4_f32(false, a, false, b1,
                                                     (short)0, acc1, false, false);
      }
    }
  }

  if (active) {
#pragma unroll
    for (int v = 0; v < 8; ++v) {         // C/D layout: M = v + 8*kh, N = m (+16)
      int row = myrow0 + v + 8 * kh;
      H1[(size_t)row * FHID + m]      = acc0[v];
      H1[(size_t)row * FHID + 16 + m] = acc1[v];
    }
  }
}

// ------------------------------------------------ edge aggregation, layer 1 (F=32)
// wave per edge: lane = feature. Edge id is wave-uniform -> scalarize index/dinv
// loads; coalesced 128B gather of h[src]; coalesced hardware f32 atomics.
__global__ __launch_bounds__(256) void agg1_kernel(const int* __restrict__ src,
                                                   const int* __restrict__ dst,
                                                   const float* __restrict__ dinv,
                                                   const float* __restrict__ H1,
                                                   float* __restrict__ agg) {
  int e = __builtin_amdgcn_readfirstlane(blockIdx.x * 8 + (threadIdx.x >> 5));
  int f = threadIdx.x & 31;
  int s = __builtin_amdgcn_readfirstlane(src[e]);
  int d = __builtin_amdgcn_readfirstlane(dst[e]);
  float norm = dinv[s] * dinv[d];
  unsafeAtomicAdd(&agg[(size_t)d * FHID + f], H1[(size_t)s * FHID + f] * norm);
}

// self-loop + bias + ReLU  (in-place into agg1 -> becomes H)
__global__ __launch_bounds__(256) void combine1_kernel(float* __restrict__ agg,
                                                       const float* __restrict__ H1,
                                                       const float* __restrict__ dinv,
                                                       const float* __restrict__ b1) {
  int i = blockIdx.x * 256 + threadIdx.x;
  if (i >= NN * FHID) return;
  int node = i >> 5;
  int f    = i & 31;
  float di = dinv[node];
  float v  = agg[i] + H1[i] * di * di + b1[f];
  agg[i] = fmaxf(v, 0.0f);
}

// ---------------------------------------------------- GEMM2: H2 = H @ W2 (WMMA f32)
__global__ __launch_bounds__(256) void gemm2_wmma(const float* __restrict__ H,
                                                  const float* __restrict__ W2,
                                                  float* __restrict__ H2) {
  __shared__ float w2c[16 * 36];   // W2 transposed: w2c[n*36 + kk] = W2[kk][n]
  const int t    = threadIdx.x;
  const int wave = t >> 5;
  const int lane = t & 31;
  const int m    = lane & 15;
  const int kh   = lane >> 4;
  const int myrow0 = blockIdx.x * 128 + wave * 16;

  if (t < 128) {
    int flat = t * 4;                 // 0..508 over 32x16 = 512 floats
    int kk = flat >> 4;               // 0..31
    int n  = flat & 15;               // 0,4,8,12
    v4f wv = *(const v4f*)(W2 + (size_t)kk * NCLS + n);
    w2c[(n + 0) * 36 + kk] = wv.x;
    w2c[(n + 1) * 36 + kk] = wv.y;
    w2c[(n + 2) * 36 + kk] = wv.z;
    w2c[(n + 3) * 36 + kk] = wv.w;
  }
  __syncthreads();
  if (myrow0 >= NN) return;           // wave-uniform

  v8f acc = {0.f,0.f,0.f,0.f,0.f,0.f,0.f,0.f};
#pragma unroll
  for (int ks = 0; ks < 8; ++ks) {    // K = 32
    int kk0 = ks * 4 + 2 * kh;
    v2f a = *(const v2f*)(H + (size_t)(myrow0 + m) * FHID + kk0);
    v2f b = *(const v2f*)(&w2c[m * 36 + kk0]);
    acc = __builtin_amdgcn_wmma_f32_16x16x4_f32(false, a, false, b,
                                                (short)0, acc, false, false);
  }
#pragma unroll
  for (int v = 0; v < 8; ++v) {
    int row = myrow0 + v + 8 * kh;
    H2[(size_t)row * NCLS + m] = acc[v];
  }
}

// ------------------------------------------------ edge aggregation, layer 2 (F=16)
__global__ __launch_bounds__(256) void agg2_kernel(const int* __restrict__ src,
                                                   const int* __restrict__ dst,
                                                   const float* __restrict__ dinv,
                                                   const float* __restrict__ H2,
                                                   float* __restrict__ agg) {
  int tid = blockIdx.x * 256 + threadIdx.x;   // NE*16 = 51.2M threads, exact grid
  int e = tid >> 4;
  int f = tid & 15;
  int s = src[e];
  int d = dst[e];
  float norm = dinv[s] * dinv[d];
  unsafeAtomicAdd(&agg[(size_t)d * NCLS + f], H2[(size_t)s * NCLS + f] * norm);
}

// self-loop + bias + log_softmax over 16 classes (16-lane shfl reductions, wave32)
__global__ __launch_bounds__(256) void out_kernel(const float* __restrict__ agg2,
                                                  const float* __restrict__ H2,
                                                  const float* __restrict__ dinv,
                                                  const float* __restrict__ b2,
                                                  float* __restrict__ out) {
  int i = blockIdx.x * 256 + threadIdx.x;     // NN*16 = 1.6M, exact grid
  int node = i >> 4;
  int j    = i & 15;
  float di = dinv[node];
  float v  = agg2[i] + H2[i] * di * di + b2[j];
  float mx = v;
#pragma unroll
  for (int off = 8; off > 0; off >>= 1)
    mx = fmaxf(mx, __shfl_xor(mx, off, 16));
  float ex = __expf(v - mx);
  float sum = ex;
#pragma unroll
  for (int off = 8; off > 0; off >>= 1)
    sum += __shfl_xor(sum, off, 16);
  out[i] = v - mx - __logf(sum);
}

// --------------------------------------------------------------------- launcher
extern "C" void kernel_launch(void* const* d_in, const int* in_sizes, int n_in,
                              void* d_out, int out_size, void* d_ws, size_t ws_size,
                              hipStream_t stream) {
  const float* x  = (const float*)d_in[0];
  const int*   ei = (const int*)d_in[1];     // [2, NE]: row0 = src, row1 = dst
  const float* W1 = (const float*)d_in[2];
  const float* b1 = (const float*)d_in[3];
  const float* W2 = (const float*)d_in[4];
  const float* b2 = (const float*)d_in[5];
  float* out = (float*)d_out;

  float* ws   = (float*)d_ws;
  float* dinv = ws;                              // NN
  float* agg1 = dinv + NN;                       // NN*32
  float* agg2 = agg1 + (size_t)NN * FHID;        // NN*16
  float* H1   = agg2 + (size_t)NN * NCLS;        // NN*32
  float* H2   = H1   + (size_t)NN * FHID;        // NN*16
  const int* src = ei;
  const int* dst = ei + NE;

  // zero deg + both aggregation buffers (contiguous 49*NN floats) every call
  int zn4 = NN * (1 + FHID + NCLS) / 4;          // 49*NN divisible by 4
  zero_ws<<<(zn4 + 255) / 256, 256, 0, stream>>>(ws, zn4);

  degree_kernel<<<NE / 256, 256, 0, stream>>>(dst, dinv);
  dinv_kernel<<<(NN + 255) / 256, 256, 0, stream>>>(dinv);

  const int mblocks = (NN / 16 + 7) / 8;         // 6250 M-tiles / 8 waves = 782
  gemm1_wmma<<<mblocks, 256, 0, stream>>>(x, W1, H1);
  agg1_kernel<<<(NE * FHID) / 256, 256, 0, stream>>>(src, dst, dinv, H1, agg1);
  combine1_kernel<<<(NN * FHID + 255) / 256, 256, 0, stream>>>(agg1, H1, dinv, b1);

  gemm2_wmma<<<mblocks, 256, 0, stream>>>(agg1 /* = H */, W2, H2);
  agg2_kernel<<<(NE * NCLS) / 256, 256, 0, stream>>>(src, dst, dinv, H2, agg2);
  out_kernel<<<(NN * NCLS) / 256, 256, 0, stream>>>(agg2, H2, dinv, b2, out);
}